// MultiHeadBDHLayer_35905926595179
// MI455X (gfx1250) — compile-verified
//
#include <hip/hip_runtime.h>
#include <math.h>

// Problem constants (from reference): B=16, T=128, N=256, H=4
#define BB 16
#define TT 128
#define NN 256
#define HH 4
#define SP (TT + 1)   // prefix-sum length per batch (S[0..T])
#define AP (TT + 2)   // padded LDS row stride (bank-conflict avoidance)

typedef float v2f __attribute__((ext_vector_type(2)));
typedef float v8f __attribute__((ext_vector_type(8)));

// CDNA5 f32 WMMA: D(16x16) = A(16x4) * B(4x16) + C, full f32 precision.
// Lane r=l&15 holds rows M=r (A) / cols N=r (B); o=l>>4 selects K pair {2o,2o+1}.
static __device__ __forceinline__ v8f wmma4(v2f a, v2f b, v8f c) {
  return __builtin_amdgcn_wmma_f32_16x16x4_f32(
      /*neg_a=*/false, a, /*neg_b=*/false, b,
      /*c_mod=*/(short)0, c, /*reuse_a=*/false, /*reuse_b=*/false);
}

// ---------------------------------------------------------------------------
// Kernel 1: exclusive prefix sums of mask per batch. S[b][t] = sum_{k<t} m_k.
// ---------------------------------------------------------------------------
__global__ void prefix_kernel(const int* __restrict__ mask, int* __restrict__ S) {
  int b = blockIdx.x;
  if (threadIdx.x == 0) {
    int* Sb = S + b * SP;
    const int* mb = mask + b * TT;
    int s = 0;
    Sb[0] = 0;
    for (int t = 0; t < TT; ++t) { s += mb[t]; Sb[t + 1] = s; }
  }
}

// ---------------------------------------------------------------------------
// Kernel 2: Gram matrices G[b] = X_b X_b^T  (T x T per batch).
// One wave per 16x16 tile; 8 waves/block cover one tile-row.
// ---------------------------------------------------------------------------
__global__ void gram_kernel(const float* __restrict__ x, float* __restrict__ G) {
  int b = blockIdx.x >> 3, ti = blockIdx.x & 7;
  int wave = threadIdx.x >> 5;          // tj tile 0..7
  int lane = threadIdx.x & 31;
  int r = lane & 15, o = lane >> 4;
  const float* Xb = x + (size_t)b * TT * NN;
  const float* ar = Xb + (size_t)(ti * 16 + r) * NN + 2 * o;   // A[t][k]
  const float* br = Xb + (size_t)(wave * 16 + r) * NN + 2 * o; // B[k][u] = X[u][k]
  v8f acc = {};
#pragma unroll 8
  for (int k = 0; k < NN; k += 4) {
    v2f a = *(const v2f*)(ar + k);
    v2f bv = *(const v2f*)(br + k);
    acc = wmma4(a, bv, acc);
  }
  float* Gb = G + (size_t)b * TT * TT;
#pragma unroll
  for (int g = 0; g < 8; ++g) {
    int t = ti * 16 + g + 8 * o; // D row mapping: VGPR g, half-wave o
    Gb[(size_t)t * TT + wave * 16 + r] = acc[g];
  }
}

// ---------------------------------------------------------------------------
// Kernel 3: base term Z[b] = X_b @ W0_b^T  (T x N per batch), shared by heads.
// ---------------------------------------------------------------------------
__global__ void __launch_bounds__(512) base_kernel(const float* __restrict__ x,
                                                   const float* __restrict__ w0,
                                                   float* __restrict__ Z) {
  int b = blockIdx.x >> 3, ti = blockIdx.x & 7;
  int wave = threadIdx.x >> 5;          // n tile 0..15
  int lane = threadIdx.x & 31;
  int r = lane & 15, o = lane >> 4;
  const float* ar = x + ((size_t)b * TT + ti * 16 + r) * NN + 2 * o;    // A[t][m]
  const float* br = w0 + ((size_t)b * NN + wave * 16 + r) * NN + 2 * o; // B[m][n]=w0[n][m]
  v8f acc = {};
#pragma unroll 8
  for (int k = 0; k < NN; k += 4) {
    v2f a = *(const v2f*)(ar + k);
    v2f bv = *(const v2f*)(br + k);
    acc = wmma4(a, bv, acc);
  }
#pragma unroll
  for (int g = 0; g < 8; ++g) {
    int t = ti * 16 + g + 8 * o;
    Z[((size_t)b * TT + t) * NN + wave * 16 + r] = acc[g];
  }
}

// ---------------------------------------------------------------------------
// Kernel 4: Y[b,t,h,:] = alpha^{S_t} * Z[b,t,:] + sum_{j<t} coef(t,j)*G[t,j]*x_j
// coef(t,j) = eta_h * m_j * alpha_h^{S_t - S_{j+1}}.
// The (Coef .* G) A-tile (16 x T) is block-uniform: built once in LDS, then
// each wave streams it into WMMA via ds_load_b64.
// ---------------------------------------------------------------------------
__global__ void __launch_bounds__(512) y_kernel(
    const float* __restrict__ x, const float* __restrict__ G,
    const float* __restrict__ Z, const int* __restrict__ S,
    const int* __restrict__ mask, const float* __restrict__ alpha,
    const float* __restrict__ eta, float* __restrict__ out_y) {
  __shared__ int sS[SP];
  __shared__ int sM[TT];
  __shared__ float sA[16][AP];   // A-tile: rows t-local, cols j (padded)
  __shared__ float sP[16];       // alpha^{S_t} per local row
  int blk = blockIdx.x;
  int ti = blk & 7;
  int h = (blk >> 3) & 3;
  int b = blk >> 5;
  int tid = threadIdx.x;
  if (tid < SP) sS[tid] = S[b * SP + tid];
  if (tid >= 256 && tid < 256 + TT) sM[tid - 256] = mask[b * TT + (tid - 256)];
  __syncthreads();

  float eh = eta[h];
  float l2a = log2f(alpha[h]);

  // Cooperative A-tile build: 2048 elements over 512 threads (coalesced G reads).
  const float* Gb = G + ((size_t)b * TT + ti * 16) * TT;
#pragma unroll
  for (int i = 0; i < 4; ++i) {
    int idx = tid + i * 512;
    int tr = idx >> 7, j = idx & 127;   // row-local t, column j
    int t = ti * 16 + tr;
    float val = 0.0f;
    if (j < t && sM[j])
      val = eh * exp2f(l2a * (float)(sS[t] - sS[j + 1])) * Gb[(size_t)tr * TT + j];
    sA[tr][j] = val;
  }
  if (tid < 16) sP[tid] = exp2f(l2a * (float)sS[ti * 16 + tid]);
  __syncthreads();

  int wave = tid >> 5;           // n tile 0..15
  int lane = tid & 31;
  int r = lane & 15, o = lane >> 4;
  const float* Xb = x + (size_t)b * TT * NN;
  int n = wave * 16 + r;

  v8f acc = {};
#pragma unroll 4
  for (int k = 0; k < TT; k += 4) {
    int j0 = k + 2 * o;
    v2f av = *(const v2f*)&sA[r][j0];            // ds_load_b64
    v2f bv;
    bv.x = Xb[(size_t)j0 * NN + n];
    bv.y = Xb[(size_t)(j0 + 1) * NN + n];
    acc = wmma4(av, bv, acc);
  }
#pragma unroll
  for (int g = 0; g < 8; ++g) {
    int tl = g + 8 * o;
    int t = ti * 16 + tl;
    float base = sP[tl] * Z[((size_t)b * TT + t) * NN + n];
    out_y[(((size_t)b * TT + t) * HH + h) * NN + n] = acc[g] + base;
  }
}

// ---------------------------------------------------------------------------
// Kernel 5: W_final[b,h] = alpha^{S_T} * W0[b] + X^T diag(d_j) X
// d_j = eta_h * m_j * alpha_h^{S_T - S_{j+1}}.
// A-tile (16 n-rows x T) = d_j * x[j,n] is block-uniform: staged in LDS.
// ---------------------------------------------------------------------------
__global__ void __launch_bounds__(512) wfin_kernel(
    const float* __restrict__ x, const float* __restrict__ w0,
    const int* __restrict__ S, const int* __restrict__ mask,
    const float* __restrict__ alpha, const float* __restrict__ eta,
    float* __restrict__ out_w) {
  __shared__ float sd[TT];
  __shared__ float sA[16][AP];
  int blk = blockIdx.x;
  int nt = blk & 15;
  int h = (blk >> 4) & 3;
  int b = blk >> 6;
  int tid = threadIdx.x;
  float eh = eta[h];
  float l2a = log2f(alpha[h]);
  int ST = S[b * SP + TT];
  if (tid < TT) {
    int m = mask[b * TT + tid];
    int cnt = ST - S[b * SP + tid + 1]; // >= 0 always
    sd[tid] = m ? eh * exp2f(l2a * (float)cnt) : 0.0f;
  }
  __syncthreads();

  // Cooperative A-tile build: sA[nr][j] = d_j * x[j, nt*16+nr].
  // Mapping idx -> (nr = idx&15, j = idx>>4): 16 consecutive threads read 16
  // contiguous floats of one x row (coalesced 64B).
  const float* Xb = x + (size_t)b * TT * NN;
#pragma unroll
  for (int i = 0; i < 4; ++i) {
    int idx = tid + i * 512;
    int nr = idx & 15, j = idx >> 4;
    sA[nr][j] = sd[j] * Xb[(size_t)j * NN + nt * 16 + nr];
  }
  __syncthreads();

  float p = exp2f(l2a * (float)ST);
  int wave = tid >> 5;           // m tile 0..15
  int lane = tid & 31;
  int r = lane & 15, o = lane >> 4;
  int m0 = wave * 16;

  v8f acc = {};
#pragma unroll 4
  for (int k = 0; k < TT; k += 4) {
    int j0 = k + 2 * o;
    v2f av = *(const v2f*)&sA[r][j0];            // ds_load_b64
    v2f bv;
    bv.x = Xb[(size_t)j0 * NN + m0 + r];         // B[j][m] = x[j][m]
    bv.y = Xb[(size_t)(j0 + 1) * NN + m0 + r];
    acc = wmma4(av, bv, acc);
  }
#pragma unroll
  for (int g = 0; g < 8; ++g) {
    int nn = nt * 16 + g + 8 * o;
    int mm = m0 + r;
    float val = acc[g] + p * w0[((size_t)b * NN + nn) * NN + mm];
    out_w[(((size_t)b * HH + h) * NN + nn) * NN + mm] = val;
  }
}

// ---------------------------------------------------------------------------
extern "C" void kernel_launch(void* const* d_in, const int* in_sizes, int n_in,
                              void* d_out, int out_size, void* d_ws, size_t ws_size,
                              hipStream_t stream) {
  const float* x     = (const float*)d_in[0]; // (B,T,N)
  const float* w0    = (const float*)d_in[1]; // (B,N,N)
  const float* alpha = (const float*)d_in[2]; // (H,)
  const float* eta   = (const float*)d_in[3]; // (H,)
  const int*   mask  = (const int*)d_in[4];   // (B,T)

  float* out_w = (float*)d_out;                         // (B,H,N,N) first
  float* out_y = out_w + (size_t)BB * HH * NN * NN;     // then (B,T,H,N)

  // Workspace: S prefix sums (8.3 KB, padded to 16 KB), G (1 MB), Z (2 MB).
  int*   S = (int*)d_ws;
  float* G = (float*)((char*)d_ws + 16384);
  float* Z = G + (size_t)BB * TT * TT;

  prefix_kernel<<<dim3(BB), dim3(32), 0, stream>>>(mask, S);
  gram_kernel<<<dim3(BB * 8), dim3(256), 0, stream>>>(x, G);
  base_kernel<<<dim3(BB * 8), dim3(512), 0, stream>>>(x, w0, Z);
  y_kernel<<<dim3(BB * HH * 8), dim3(512), 0, stream>>>(x, G, Z, S, mask, alpha, eta, out_y);
  wfin_kernel<<<dim3(BB * HH * 16), dim3(512), 0, stream>>>(x, w0, S, mask, alpha, eta, out_w);
}